// DiffusionGraphConv_85023172592643
// MI455X (gfx1250) — compile-verified
//
#include <hip/hip_runtime.h>

#define NNODES 50000
#define DFEAT 64
#define OFEAT 64
#define KM 5
#define LDSTRIDE 66   // 64 + 2 pad: lane-to-lane LDS addr stride 66*4B -> banks 2 apart

typedef float v2f __attribute__((ext_vector_type(2)));
typedef float v8f __attribute__((ext_vector_type(8)));

// ---------------- elementwise helpers ----------------

__global__ void zero_f32(float* __restrict__ p, int n) {
  int i = blockIdx.x * blockDim.x + threadIdx.x;
  if (i < n) p[i] = 0.0f;
}

__global__ void negcopy_f32(const float* __restrict__ x, float* __restrict__ y, int n) {
  int i = blockIdx.x * blockDim.x + threadIdx.x;
  if (i < n) y[i] = -x[i];
}

// ---------------- SpMM: y[r,:] += scale * v * x[c,:]  (64 features) ----------------
// One thread handles (edge, 4 consecutive features). Consecutive 16 threads share an
// edge -> coalesced float4 gather from x (L2-resident: 12.8MB per-b slab) and
// contiguous global_atomic_add_f32 scatter.

__global__ void spmm_atomic(const int* __restrict__ rows, const int* __restrict__ cols,
                            const float* __restrict__ vals,
                            const float* __restrict__ xsrc, float* __restrict__ y,
                            float scale, int E) {
  long long tid = (long long)blockIdx.x * blockDim.x + threadIdx.x;
  long long e = tid >> 4;
  if (e >= E) return;
  int f4 = ((int)tid & 15) << 2;
  int c = cols[e];
  int r = rows[e];
  float v = vals[e] * scale;
  float4 xv = *(const float4*)(xsrc + (size_t)c * DFEAT + f4);
  float* yp = y + (size_t)r * DFEAT + f4;
  unsafeAtomicAdd(yp + 0, v * xv.x);
  unsafeAtomicAdd(yp + 1, v * xv.y);
  unsafeAtomicAdd(yp + 2, v * xv.z);
  unsafeAtomicAdd(yp + 3, v * xv.w);
}

// ---------------- final GEMM via V_WMMA_F32_16X16X4_F32 ----------------
// out[m, o] = bias[o] + sum_k sum_d src_k[m, d] * W[d*5+k, o]
// One wave computes a 16(M) x 64(N) tile with 4 v8f accumulators.
// A frag (16x4 f32): lane<16 -> row M=l16, K={0,1}; lane>=16 -> M=l16, K={2,3}.
// B frag (4x16 f32): VGPR0 = rows K=0 (lanes 0-15) / K=2 (lanes 16-31), VGPR1 = K=1/K=3.
// Weight staged in LDS transposed [k][o][d] so the per-lane B pair (d, d+1) is one ds_load_b64.

__global__ void __launch_bounds__(256)
gemm_wmma_f32(const float* __restrict__ s0, const float* __restrict__ s1,
              const float* __restrict__ s2, const float* __restrict__ s3,
              const float* __restrict__ s4, const float* __restrict__ W,
              const float* __restrict__ bias, float* __restrict__ out) {
  extern __shared__ float lds[];  // KM * OFEAT * LDSTRIDE floats

  // cooperative transpose-load of weight: lds[(k*64 + o)*66 + d] = W[(d*5+k)*64 + o]
  for (int idx = threadIdx.x; idx < KM * DFEAT * OFEAT; idx += blockDim.x) {
    int row = idx >> 6;   // d*5 + k
    int o   = idx & 63;
    int d   = row / KM;
    int k   = row % KM;
    lds[(k * OFEAT + o) * LDSTRIDE + d] = W[idx];
  }
  __syncthreads();

  int wave = threadIdx.x >> 5;
  int lane = threadIdx.x & 31;
  int half = lane >> 4;   // 0: lanes 0-15, 1: lanes 16-31
  int l16  = lane & 15;

  int m0 = (blockIdx.x * 8 + wave) * 16;
  if (m0 >= NNODES) return;   // wave-uniform; active waves keep EXEC all-1s

  v8f acc[4];
  for (int nt = 0; nt < 4; ++nt) {
    float bv = bias[nt * 16 + l16];
    for (int j = 0; j < 8; ++j) acc[nt][j] = bv;
  }

  const float* srcs[KM] = {s0, s1, s2, s3, s4};
#pragma unroll
  for (int k = 0; k < KM; ++k) {
    const float* arow  = srcs[k] + (size_t)(m0 + l16) * DFEAT + half * 2;
    const float* bbase = lds + (size_t)(k * OFEAT) * LDSTRIDE + half * 2;
#pragma unroll
    for (int kk = 0; kk < 16; ++kk) {           // K = 64 in steps of 4
      v2f a = *(const v2f*)(arow + kk * 4);
#pragma unroll
      for (int nt = 0; nt < 4; ++nt) {
        v2f b = *(const v2f*)(bbase + (nt * 16 + l16) * LDSTRIDE + kk * 4);
        acc[nt] = __builtin_amdgcn_wmma_f32_16x16x4_f32(
            /*neg_a=*/false, a, /*neg_b=*/false, b,
            /*c_mod=*/(short)0, acc[nt], /*reuse_a=*/false, /*reuse_b=*/false);
      }
    }
  }

  // C/D layout: VGPR j -> M = j (lanes 0-15) / M = j+8 (lanes 16-31), N = l16
  size_t outbase = (size_t)(m0 + half * 8) * OFEAT;
#pragma unroll
  for (int nt = 0; nt < 4; ++nt)
#pragma unroll
    for (int j = 0; j < 8; ++j)
      out[outbase + (size_t)j * OFEAT + nt * 16 + l16] = acc[nt][j];
}

// ---------------- launcher ----------------

extern "C" void kernel_launch(void* const* d_in, const int* in_sizes, int n_in,
                              void* d_out, int out_size, void* d_ws, size_t ws_size,
                              hipStream_t stream) {
  const float* x     = (const float*)d_in[0];
  const int*   rows0 = (const int*)d_in[1];
  const int*   cols0 = (const int*)d_in[2];
  const float* vals0 = (const float*)d_in[3];
  const int*   rows1 = (const int*)d_in[4];
  const int*   cols1 = (const int*)d_in[5];
  const float* vals1 = (const float*)d_in[6];
  const float* W     = (const float*)d_in[7];
  const float* bias  = (const float*)d_in[8];
  float* out = (float*)d_out;
  int E = in_sizes[1];

  const size_t mat = (size_t)NNODES * DFEAT;  // per-b feature slab (50000 x 64)
  float* s1 = (float*)d_ws;       // A0 * x0
  float* s2 = s1 + mat;           // 2*A0*s1 - x0
  float* s3 = s2 + mat;           // A1 * x0
  float* s4 = s3 + mat;           // 2*A1*s3 - x0

  int eblocks = (int)(((long long)E * 16 + 255) / 256);
  int nblocks = (int)((mat + 255) / 256);
  int gemmBlocks = (NNODES / 16 + 7) / 8;     // 3125 tiles / 8 waves per block
  size_t ldsBytes = (size_t)KM * OFEAT * LDSTRIDE * sizeof(float);

  for (int b = 0; b < 8; ++b) {
    const float* xb = x + (size_t)b * mat;    // x0 slice for batch b: [n, d] contiguous
    float* ob = out + (size_t)b * mat;        // out slice: [n, o] contiguous

    // support 0: x1 = A0 x0 ; x2 = 2 A0 x1 - x0 (init accumulator with -x0)
    zero_f32<<<nblocks, 256, 0, stream>>>(s1, (int)mat);
    spmm_atomic<<<eblocks, 256, 0, stream>>>(rows0, cols0, vals0, xb, s1, 1.0f, E);
    negcopy_f32<<<nblocks, 256, 0, stream>>>(xb, s2, (int)mat);
    spmm_atomic<<<eblocks, 256, 0, stream>>>(rows0, cols0, vals0, s1, s2, 2.0f, E);

    // support 1
    zero_f32<<<nblocks, 256, 0, stream>>>(s3, (int)mat);
    spmm_atomic<<<eblocks, 256, 0, stream>>>(rows1, cols1, vals1, xb, s3, 1.0f, E);
    negcopy_f32<<<nblocks, 256, 0, stream>>>(xb, s4, (int)mat);
    spmm_atomic<<<eblocks, 256, 0, stream>>>(rows1, cols1, vals1, s3, s4, 2.0f, E);

    // out[b] = [x0 | s1 | s2 | s3 | s4] (interleaved d*5+k) @ W + bias
    gemm_wmma_f32<<<gemmBlocks, 256, ldsBytes, stream>>>(xb, s1, s2, s3, s4, W, bias, ob);
  }
}